// Normalize_51969104282122
// MI455X (gfx1250) — compile-verified
//
#include <hip/hip_runtime.h>
#include <hip/hip_bf16.h>

typedef float v4f __attribute__((ext_vector_type(4)));

#define EPS 1e-8f

constexpr int SLICE   = 128 * 128;                // 16384 floats per (i,j,k) slice
constexpr int THREADS = 256;                      // 8 wave32 waves
constexpr int VPT     = SLICE / (THREADS * 4);    // 16 float4 chunks per thread
constexpr int CHUNK_BYTES = THREADS * 16;         // 4096 B stride between chunks

__global__ __launch_bounds__(THREADS)
void Normalize_51969104282122_kernel(const float* __restrict__ x,
                                     float* __restrict__ out) {
    __shared__ __align__(16) float tile[SLICE];   // 64 KB slice staging
    __shared__ float swmin[8];
    __shared__ float swmax[8];

    const int tid = threadIdx.x;
    const long long slice = blockIdx.x;
    const float* __restrict__ src = x   + slice * (long long)SLICE;
    float*       __restrict__ dst = out + slice * (long long)SLICE;

    // ---- Stage slice into LDS via gfx1250 async global->LDS copies ----
    // saddr form: global addr = SGPR base + lane_off + imm_offset, and the
    // same imm_offset is applied to the LDS destination address, so all 16
    // issues share one base -> zero per-chunk VALU address math.
    const unsigned lane_off = (unsigned)tid * 16u;                       // per-lane byte offset
    const unsigned lds_base = (unsigned)(uintptr_t)tile + lane_off;      // low 32b = LDS byte offset
    const unsigned long long saddr = (unsigned long long)(uintptr_t)src;
    #pragma unroll
    for (int i = 0; i < VPT; ++i) {
        asm volatile("global_load_async_to_lds_b128 %0, %1, %2 offset:%3"
                     :: "v"(lds_base), "v"(lane_off), "s"(saddr),
                        "n"(i * CHUNK_BYTES)
                     : "memory");
    }
    asm volatile("s_wait_asynccnt 0" ::: "memory");   // my wave's async copies done
    __syncthreads();                                  // all waves' copies done

    // ---- Read slice from LDS ONCE into registers; local min/max ----
    v4f d[VPT];
    float vmin =  3.402823466e38f;
    float vmax = -3.402823466e38f;
    #pragma unroll
    for (int i = 0; i < VPT; ++i) {
        const int v = i * THREADS + tid;
        d[i] = *(const v4f*)&tile[v * 4];
        vmin = fminf(vmin, fminf(fminf(d[i].x, d[i].y), fminf(d[i].z, d[i].w)));
        vmax = fmaxf(vmax, fmaxf(fmaxf(d[i].x, d[i].y), fmaxf(d[i].z, d[i].w)));
    }

    // ---- wave32 butterfly reduction ----
    #pragma unroll
    for (int off = 16; off >= 1; off >>= 1) {
        vmin = fminf(vmin, __shfl_xor(vmin, off, 32));
        vmax = fmaxf(vmax, __shfl_xor(vmax, off, 32));
    }

    // ---- cross-wave reduction through LDS (8 waves) ----
    const int wave = tid >> 5;
    const int lane = tid & 31;
    if (lane == 0) { swmin[wave] = vmin; swmax[wave] = vmax; }
    __syncthreads();

    float bmin = swmin[0];
    float bmax = swmax[0];
    #pragma unroll
    for (int w = 1; w < 8; ++w) {
        bmin = fminf(bmin, swmin[w]);
        bmax = fmaxf(bmax, swmax[w]);
    }
    const float scale = 1.0f / (bmax - bmin + EPS);

    // ---- Normalize from registers, stream out with non-temporal stores ----
    #pragma unroll
    for (int i = 0; i < VPT; ++i) {
        const int v = i * THREADS + tid;
        v4f r;
        r.x = (d[i].x - bmin) * scale;
        r.y = (d[i].y - bmin) * scale;
        r.z = (d[i].z - bmin) * scale;
        r.w = (d[i].w - bmin) * scale;
        __builtin_nontemporal_store(r, (v4f*)(dst + v * 4));
    }
}

extern "C" void kernel_launch(void* const* d_in, const int* in_sizes, int n_in,
                              void* d_out, int out_size, void* d_ws, size_t ws_size,
                              hipStream_t stream) {
    const float* x = (const float*)d_in[0];
    float* out = (float*)d_out;
    const int n_slices = out_size / SLICE;   // 6144 for (16,12,32,128,128)
    Normalize_51969104282122_kernel<<<n_slices, THREADS, 0, stream>>>(x, out);
}